// BCE_Loss_31507880083477
// MI455X (gfx1250) — compile-verified
//
#include <hip/hip_runtime.h>
#include <hip/hip_bf16.h>
#include <math.h>

typedef __attribute__((ext_vector_type(2))) float v2f;
typedef __attribute__((ext_vector_type(4))) float v4f;
typedef __attribute__((ext_vector_type(8))) float v8f;
typedef __attribute__((ext_vector_type(4))) int   v4i;

// ---------------------------------------------------------------------------
// Wave32 sum-reduction through the matrix pipe (v_wmma_f32_16x16x4_f32).
// A-matrix (16x4 f32, 2 VGPRs/lane): VGPR0 = per-lane value v, VGPR1 = 0.
//   => A(m,0)=v[m], A(m,1)=0, A(m,2)=v[m+16], A(m,3)=0   (m = lane%16)
// B-matrix = all ones, so D(m,n) = v[m] + v[m+16] regardless of B layout.
// D layout: VGPR r, lanes 0-15 hold D(r,n); lanes 16-31 hold D(r+8,n).
// Summing the 8 D VGPRs:
//   lanes 0-15  -> sum over source lanes {0..7, 16..23}
//   lanes 16-31 -> sum over source lanes {8..15, 24..31}
// i.e. lanes 0 and 16 together hold the full wave sum split in two parts;
// the caller adds the two halves during the LDS cross-wave stage (no shuffle).
// EXEC must be all 1s here (called after loop reconvergence).
// ---------------------------------------------------------------------------
__device__ __forceinline__ float wave_reduce_wmma_half(float v) {
  v2f a; a[0] = v;    a[1] = 0.0f;
  v2f b; b[0] = 1.0f; b[1] = 1.0f;
  v8f c = {};
  v8f d = __builtin_amdgcn_wmma_f32_16x16x4_f32(
      /*neg_a=*/false, a, /*neg_b=*/false, b,
      /*c_mod=*/(short)0, c, /*reuse_a=*/false, /*reuse_b=*/false);
  return d[0] + d[1] + d[2] + d[3] + d[4] + d[5] + d[6] + d[7];
}

// Branchless focal-weighted BCE-with-logits element (matches reference):
//   ip = one-hot indicator (1.0 for the target class, else 0.0)
//   weight = (x - ip)^2          [(1-x)^2 if ip=1, x^2 if ip=0]
//   bce    = max(x,0) - ip*x + log1p(exp(-|x|))
__device__ __forceinline__ float focal_elem(float x, float ip) {
  float l   = fmaxf(x, 0.0f);
  float s   = log1pf(__expf(-fabsf(x)));
  float dxi = x - ip;
  float bce = l + s - ip * x;
  return dxi * dxi * bce;
}

// per-anchor contribution: 3 logits + int target -> branchless one-hot
__device__ __forceinline__ float focal_anchor(float x0, float x1, float x2, int tv) {
  const float ip0 = (tv == 1) ? 1.0f : 0.0f;   // class id 1 -> column 0
  const float ip1 = (tv == 3) ? 1.0f : 0.0f;   // class id 3 -> column 1
  const float ip2 = 1.0f - ip0 - ip1;          // else        -> column 2
  return focal_elem(x0, ip0) + focal_elem(x1, ip1) + focal_elem(x2, ip2);
}

// cross-wave stage shared by both kernels: lanes 0 and 16 of each of the
// 8 waves deposit their half-sums; thread 0 adds the 16 values.
__device__ __forceinline__ void block_reduce_store(float acc, float* dst) {
  const float half = wave_reduce_wmma_half(acc);

  __shared__ float sm[16];
  const int lane = threadIdx.x & 31;
  const int wave = threadIdx.x >> 5;
  if ((lane & 15) == 0) sm[wave * 2 + (lane >> 4)] = half;
  __syncthreads();
  if (threadIdx.x == 0) {
    float b = 0.0f;
#pragma unroll
    for (int i = 0; i < 16; ++i) b += sm[i];
    dst[0] = b;
  }
}

__launch_bounds__(256)
__global__ void focal_partial_kernel(const float* __restrict__ pred,
                                     const int*   __restrict__ targ,
                                     float*       __restrict__ partial,
                                     int n_anchors) {
  const int tid      = blockIdx.x * blockDim.x + threadIdx.x;
  const int nthreads = gridDim.x * blockDim.x;
  const int nchunks  = n_anchors >> 2;                 // 4 anchors per chunk

  const v4f* __restrict__ pred4 = (const v4f*)pred;    // 12 floats = 3 x b128
  const v4i* __restrict__ targ4 = (const v4i*)targ;

  float acc = 0.0f;

  for (int ci = tid; ci < nchunks; ci += nthreads) {
    // prefetch next grid-stride chunk (speculative: OOB silently dropped)
    __builtin_prefetch(&pred4[(size_t)(ci + nthreads) * 3], 0, 0);

    // single-pass stream: non-temporal 128-bit loads
    const v4i tg = __builtin_nontemporal_load(&targ4[ci]);
    const v4f p0 = __builtin_nontemporal_load(&pred4[(size_t)ci * 3 + 0]);
    const v4f p1 = __builtin_nontemporal_load(&pred4[(size_t)ci * 3 + 1]);
    const v4f p2 = __builtin_nontemporal_load(&pred4[(size_t)ci * 3 + 2]);

    acc += focal_anchor(p0[0], p0[1], p0[2], tg[0]);
    acc += focal_anchor(p0[3], p1[0], p1[1], tg[1]);
    acc += focal_anchor(p1[2], p1[3], p2[0], tg[2]);
    acc += focal_anchor(p2[1], p2[2], p2[3], tg[3]);
  }

  // scalar tail (n_anchors % 4); reconverges before the WMMA reduction
  if (tid == 0) {
#pragma unroll 1
    for (int a = nchunks << 2; a < n_anchors; ++a) {
      acc += focal_anchor(pred[(size_t)a * 3 + 0],
                          pred[(size_t)a * 3 + 1],
                          pred[(size_t)a * 3 + 2], targ[a]);
    }
  }

  block_reduce_store(acc, &partial[blockIdx.x]);
}

__launch_bounds__(256)
__global__ void focal_final_kernel(const float* __restrict__ partial,
                                   int n_partials,
                                   float* __restrict__ out,
                                   float inv_count) {
  float acc = 0.0f;
#pragma unroll 1
  for (int i = threadIdx.x; i < n_partials; i += blockDim.x) acc += partial[i];

  __shared__ float blocksum;
  block_reduce_store(acc, &blocksum);
  __syncthreads();
  if (threadIdx.x == 0) out[0] = blocksum * inv_count;
}

extern "C" void kernel_launch(void* const* d_in, const int* in_sizes, int n_in,
                              void* d_out, int out_size, void* d_ws, size_t ws_size,
                              hipStream_t stream) {
  const float* pred = (const float*)d_in[0];   // [A, 3] f32
  const int*   targ = (const int*)d_in[1];     // [A] int32
  float*       out  = (float*)d_out;           // scalar f32
  float*       part = (float*)d_ws;            // block partials (8 KB)

  const int n_anchors = in_sizes[1];

  const int NT = 256;   // 8 waves / block
  const int NB = 2048;  // exactly 4 chunk-iterations per thread at A = 8M

  focal_partial_kernel<<<NB, NT, 0, stream>>>(pred, targ, part, n_anchors);

  const float inv_count = 1.0f / (3.0f * (float)n_anchors);
  focal_final_kernel<<<1, NT, 0, stream>>>(part, NB, out, inv_count);
}